// Grid_88914412961874
// MI455X (gfx1250) — compile-verified
//
#include <hip/hip_runtime.h>
#include <stdint.h>

// Problem constants (from reference)
#define GH 1024
#define GW 1024
#define GC 32

// Native 128-bit float vector (works with __builtin_nontemporal_store,
// unlike HIP's float4 class type)
typedef float f32x4 __attribute__((ext_vector_type(4)));
typedef float f32x2 __attribute__((ext_vector_type(2)));

// Address-space-qualified pointer typedefs
typedef __attribute__((address_space(1))) int  as1_int;
typedef __attribute__((address_space(3))) int  as3_int;

// ---------------------------------------------------------------------------
// Detect CDNA5 async global->LDS builtins (guarded so compile never fails).
// ---------------------------------------------------------------------------
#if defined(__has_builtin)
#if __has_builtin(__builtin_amdgcn_global_load_async_to_lds_b32)
#define HAVE_ASYNC_LDS 1
#endif
#if __has_builtin(__builtin_amdgcn_s_wait_asynccnt)
#define HAVE_WAIT_ASYNC 1
#endif
#endif

__device__ __forceinline__ void wait_asynccnt0() {
#if defined(HAVE_WAIT_ASYNC)
    __builtin_amdgcn_s_wait_asynccnt(0);
#elif defined(__gfx1250__)
    asm volatile("s_wait_asynccnt 0" ::: "memory");
#endif
}

// ---------------------------------------------------------------------------
// Pass 1: repack params [C,H,W] -> [H,W,C] so each (y,x) sample point's 32
// channels are one contiguous, 128B-aligned cache line. 32x32 LDS tile
// transpose; reads and writes both fully coalesced (128B per wave).
// On CDNA5 the global->LDS leg uses GLOBAL_LOAD_ASYNC_TO_LDS (ASYNCcnt).
// ---------------------------------------------------------------------------
__global__ __launch_bounds__(1024) void repack_kernel(const float* __restrict__ src,
                                                      float* __restrict__ dst) {
    __shared__ float tile[32 * 33];  // +1 padding column: conflict-free
    const int x0  = blockIdx.x << 5;
    const int y   = blockIdx.y;
    const int tid = threadIdx.x;

    // phase 1: lanes stride along W (contiguous in src)
    const int w = tid & 31;
    const int c = tid >> 5;
    const size_t src_idx = (size_t)c * (GH * GW) + (size_t)y * GW + x0 + w;
    const int lds_off = w * 33 + c;

#if defined(HAVE_ASYNC_LDS) && defined(__gfx1250__)
    {
        // (global src ptr, LDS dst ptr, imm offset, imm cpol)
        as1_int* gp = (as1_int*)(src + src_idx);
        as3_int* lp = (as3_int*)(&tile[lds_off]);
        __builtin_amdgcn_global_load_async_to_lds_b32(gp, lp, 0, 0);
        wait_asynccnt0();
    }
#else
    tile[lds_off] = src[src_idx];
#endif
    __syncthreads();

    // phase 2: lanes stride along C (contiguous in dst)
    const int c2 = tid & 31;
    const int w2 = tid >> 5;
    dst[((size_t)y * GW + x0 + w2) * GC + c2] = tile[w2 * 33 + c2];
}

// ---------------------------------------------------------------------------
// Pass 2: bilinear sample from repacked [H,W,C] layout.
// Thread t -> point p = t>>3, channel quad cq = t&7 (channels 4*cq..4*cq+3).
// Per point: 4 corner float4 (B128) loads -- each 8-lane group covers one
// 128B line -- and 1 float4 non-temporal store (keeps the 128MB param table
// resident in the 192MB L2 instead of letting the 256MB output stream evict
// it).
// ---------------------------------------------------------------------------
__global__ __launch_bounds__(256) void sample_packed_kernel(const float* __restrict__ coord,
                                                            const float* __restrict__ rp,
                                                            float* __restrict__ out,
                                                            int npts) {
    const int t  = blockIdx.x * blockDim.x + threadIdx.x;
    const int p  = t >> 3;
    const int cq = t & 7;
    if (p >= npts) return;

    const f32x2 xy = ((const f32x2*)coord)[p];

    // align_corners=True mapping, mirroring reference arithmetic
    const float ix = (xy.x + 1.0f) * 0.5f * (float)(GW - 1);
    const float iy = (xy.y + 1.0f) * 0.5f * (float)(GH - 1);
    const float ix0f = floorf(ix), iy0f = floorf(iy);
    const float ix1f = ix0f + 1.0f, iy1f = iy0f + 1.0f;

    const float w_nw = (ix1f - ix) * (iy1f - iy);
    const float w_ne = (ix - ix0f) * (iy1f - iy);
    const float w_sw = (ix1f - ix) * (iy - iy0f);
    const float w_se = (ix - ix0f) * (iy - iy0f);

    const int ix0 = min(max((int)ix0f, 0), GW - 1);
    const int ix1 = min(max((int)ix0f + 1, 0), GW - 1);
    const int iy0 = min(max((int)iy0f, 0), GH - 1);
    const int iy1 = min(max((int)iy0f + 1, 0), GH - 1);

    const f32x4* nw = (const f32x4*)(rp + ((size_t)iy0 * GW + ix0) * GC) + cq;
    const f32x4* ne = (const f32x4*)(rp + ((size_t)iy0 * GW + ix1) * GC) + cq;
    const f32x4* sw = (const f32x4*)(rp + ((size_t)iy1 * GW + ix0) * GC) + cq;
    const f32x4* se = (const f32x4*)(rp + ((size_t)iy1 * GW + ix1) * GC) + cq;

    const f32x4 a = *nw, b = *ne, cc = *sw, d = *se;

    f32x4 r;
    r.x = a.x * w_nw + b.x * w_ne + cc.x * w_sw + d.x * w_se;
    r.y = a.y * w_nw + b.y * w_ne + cc.y * w_sw + d.y * w_se;
    r.z = a.z * w_nw + b.z * w_ne + cc.z * w_sw + d.z * w_se;
    r.w = a.w * w_nw + b.w * w_ne + cc.w * w_sw + d.w * w_se;

    f32x4* op = (f32x4*)(out + (size_t)p * GC) + cq;
    __builtin_nontemporal_store(r, op);
}

// ---------------------------------------------------------------------------
// Fallback (if workspace can't hold the 128MB repack): direct gather from the
// original [C,H,W] layout; lane = channel.
// ---------------------------------------------------------------------------
__global__ __launch_bounds__(256) void sample_direct_kernel(const float* __restrict__ coord,
                                                            const float* __restrict__ params,
                                                            float* __restrict__ out,
                                                            int npts) {
    const long long t = (long long)blockIdx.x * blockDim.x + threadIdx.x;
    const int p = (int)(t >> 5);
    const int c = (int)(t & 31);
    if (p >= npts) return;

    const f32x2 xy = ((const f32x2*)coord)[p];
    const float ix = (xy.x + 1.0f) * 0.5f * (float)(GW - 1);
    const float iy = (xy.y + 1.0f) * 0.5f * (float)(GH - 1);
    const float ix0f = floorf(ix), iy0f = floorf(iy);
    const float ix1f = ix0f + 1.0f, iy1f = iy0f + 1.0f;

    const float w_nw = (ix1f - ix) * (iy1f - iy);
    const float w_ne = (ix - ix0f) * (iy1f - iy);
    const float w_sw = (ix1f - ix) * (iy - iy0f);
    const float w_se = (ix - ix0f) * (iy - iy0f);

    const int ix0 = min(max((int)ix0f, 0), GW - 1);
    const int ix1 = min(max((int)ix0f + 1, 0), GW - 1);
    const int iy0 = min(max((int)iy0f, 0), GH - 1);
    const int iy1 = min(max((int)iy0f + 1, 0), GH - 1);

    const float* pc = params + (size_t)c * (GH * GW);
    const float v_nw = pc[(size_t)iy0 * GW + ix0];
    const float v_ne = pc[(size_t)iy0 * GW + ix1];
    const float v_sw = pc[(size_t)iy1 * GW + ix0];
    const float v_se = pc[(size_t)iy1 * GW + ix1];

    out[(size_t)p * GC + c] = v_nw * w_nw + v_ne * w_ne + v_sw * w_sw + v_se * w_se;
}

extern "C" void kernel_launch(void* const* d_in, const int* in_sizes, int n_in,
                              void* d_out, int out_size, void* d_ws, size_t ws_size,
                              hipStream_t stream) {
    const float* coord  = (const float*)d_in[0];   // [N,2] fp32
    const float* params = (const float*)d_in[1];   // [1,32,1024,1024] fp32
    float* out = (float*)d_out;                    // [N,32] fp32
    const int npts = in_sizes[0] / 2;

    const size_t repack_bytes = (size_t)GH * GW * GC * sizeof(float);  // 128 MB
    if (ws_size >= repack_bytes) {
        float* rp = (float*)d_ws;
        dim3 grid_rp(GW / 32, GH);
        repack_kernel<<<grid_rp, 1024, 0, stream>>>(params, rp);

        const long long threads = (long long)npts * 8;
        const int blocks = (int)((threads + 255) / 256);
        sample_packed_kernel<<<blocks, 256, 0, stream>>>(coord, rp, out, npts);
    } else {
        const long long threads = (long long)npts * 32;
        const int blocks = (int)((threads + 255) / 256);
        sample_direct_kernel<<<blocks, 256, 0, stream>>>(coord, params, out, npts);
    }
}